// SDF_23235773071365
// MI455X (gfx1250) — compile-verified
//
#include <hip/hip_runtime.h>
#include <math.h>

// CDNA5 / gfx1250 — wave32, WMMA f32 16x16x4.
typedef __attribute__((ext_vector_type(2))) float v2f;
typedef __attribute__((ext_vector_type(8))) float v8f;

#define WPB   8   // waves per 256-thread block
#define SPLIT 2   // waves cooperating on one 16-query tile (cloud split)
#define KNN   8
#define EPS   1e-12f

// Score = |c|^2 - 2 c.p via V_WMMA_F32_16X16X4_F32 with
// A-row = [-2cx,-2cy,-2cz,|c|^2], B-col = [px,py,pz,1]  (K=4 folds the bias).
// D layout (16x16 f32 C/D): lane l, VGPR r -> M = r + 8*(l>=16), N = l&15.
// Each lane keeps a private top-8 for query (l&15). A query's candidates live
// in 2*SPLIT lanes (two half-waves x SPLIT segment waves); merged via LDS.
__global__ __launch_bounds__(256) void sdf_knn_wmma(
    const float* __restrict__ pts,      // [Q][3] flattened (B*N queries)
    const float* __restrict__ cloud,    // [M][3]
    const float* __restrict__ normals,  // [M][3]
    const float* __restrict__ conf,     // [M]
    float* __restrict__ out,            // [Q]
    int Q, int M)
{
    __shared__ float lds_s[WPB][KNN][32];
    __shared__ int   lds_i[WPB][KNN][32];

    const int lane = threadIdx.x & 31;
    const int wv   = threadIdx.x >> 5;
    const int half = lane >> 4;      // 0 = lanes 0-15, 1 = lanes 16-31
    const int n    = lane & 15;      // query column within tile

    const int  Qtiles = Q >> 4;
    const int  gw     = blockIdx.x * WPB + wv;   // global wave id
    const int  qtile  = gw / SPLIT;              // 16-query tile
    const int  seg    = gw % SPLIT;              // cloud segment
    const bool active = (qtile < Qtiles);        // wave-uniform
    const int  qbase  = (active ? qtile : 0) << 4;
    const int  q      = qbase + n;

    // ---- B matrix (4x16): column n = [px, py, pz, 1].
    // Layout (mirror of documented A): VGPR0 = K0 row (lanes 0-15) / K2 row
    // (lanes 16-31); VGPR1 = K1 / K3.
    const float px = pts[q * 3 + 0];
    const float py = pts[q * 3 + 1];
    const float pz = pts[q * 3 + 2];
    v2f Bm;
    Bm.x = half ? pz   : px;
    Bm.y = half ? 1.0f : py;

    // ---- per-lane top-K list, ascending (ks[KNN-1] = current worst)
    float ks[KNN];
    int   ki[KNN];
#pragma unroll
    for (int j = 0; j < KNN; ++j) { ks[j] = 3.0e38f; ki[j] = 0; }

    const int fullTiles   = M >> 4;                       // all-valid tiles
    const int tilesPerSeg = (fullTiles + SPLIT - 1) / SPLIT;
    int t0   = seg * tilesPerSeg;
    int tend = t0 + tilesPerSeg;
    if (tend > fullTiles) tend = fullTiles;
    if (!active) { t0 = 0; tend = 0; }

    // ---- pipelined preload of tile t0; unsigned byte-offset march keeps
    // addressing at add+min per iteration (saddr + 32-bit voffset form).
    const unsigned offMax = (unsigned)(M - 1) * 12u;
    unsigned off  = ((unsigned)(t0 * 16 + n)) * 12u;
    unsigned offc = (off < offMax) ? off : offMax;
    const char* cbase = (const char*)cloud;
    float cx, cy, cz;
    {
        const float* p = (const float*)(cbase + offc);
        cx = p[0]; cy = p[1]; cz = p[2];
    }

    for (int t = t0; t < tend; ++t) {
        // ---- A matrix (16x4 f32, ISA layout): lane l holds row M = l&15;
        // lanes 0-15 carry K=0,1 and lanes 16-31 carry K=2,3.
        v2f A;
        A.x = half ? (-2.0f * cz) : (-2.0f * cx);
        A.y = half ? (cx * cx + cy * cy + cz * cz) : (-2.0f * cy);

        // ---- prefetch tile t+1 (clamped byte offset; wait lands next iter)
        off += 16u * 12u;
        offc = (off < offMax) ? off : offMax;
        const float* p = (const float*)(cbase + offc);
        float ncx = p[0];
        float ncy = p[1];
        float ncz = p[2];

        v8f c = {0.f, 0.f, 0.f, 0.f, 0.f, 0.f, 0.f, 0.f};
        v8f d = __builtin_amdgcn_wmma_f32_16x16x4_f32(
            /*neg_a=*/false, A, /*neg_b=*/false, Bm,
            /*c_mod=*/(short)0, c, /*reuse_a=*/false, /*reuse_b=*/false);

        // ---- min-tree prefilter: one branch per tile, rarely taken
        float m01 = fminf(d[0], d[1]);
        float m23 = fminf(d[2], d[3]);
        float m45 = fminf(d[4], d[5]);
        float m67 = fminf(d[6], d[7]);
        float mall = fminf(fminf(m01, m23), fminf(m45, m67));

        if (mall < ks[KNN - 1]) {
            const int mbase = t * 16 + (half << 3);
#pragma unroll
            for (int r = 0; r < KNN; ++r) {
                float s = d[r];
                if (s < ks[KNN - 1]) {
                    ks[KNN - 1] = s; ki[KNN - 1] = mbase + r;
#pragma unroll
                    for (int j = KNN - 1; j > 0; --j) {
                        if (ks[j] < ks[j - 1]) {
                            float ts = ks[j]; ks[j] = ks[j - 1]; ks[j - 1] = ts;
                            int   ti = ki[j]; ki[j] = ki[j - 1]; ki[j - 1] = ti;
                        }
                    }
                }
            }
        }
        cx = ncx; cy = ncy; cz = ncz;
    }

    // ---- tail tile (M % 16 != 0), handled by last segment only.
    // Uniform condition: EXEC stays full for the WMMA.
    if (active && seg == SPLIT - 1 && (M & 15)) {
        v2f A;
        A.x = half ? (-2.0f * cz) : (-2.0f * cx);
        A.y = half ? (cx * cx + cy * cy + cz * cz) : (-2.0f * cy);
        v8f c = {0.f, 0.f, 0.f, 0.f, 0.f, 0.f, 0.f, 0.f};
        v8f d = __builtin_amdgcn_wmma_f32_16x16x4_f32(
            false, A, false, Bm, (short)0, c, false, false);
        const int mbase = fullTiles * 16 + (half << 3);
#pragma unroll
        for (int r = 0; r < KNN; ++r) {
            float s = d[r];
            int   m = mbase + r;
            if (m < M && s < ks[KNN - 1]) {
                ks[KNN - 1] = s; ki[KNN - 1] = m;
#pragma unroll
                for (int j = KNN - 1; j > 0; --j) {
                    if (ks[j] < ks[j - 1]) {
                        float ts = ks[j]; ks[j] = ks[j - 1]; ks[j - 1] = ts;
                        int   ti = ki[j]; ki[j] = ki[j - 1]; ki[j - 1] = ti;
                    }
                }
            }
        }
    }

    // ---- publish per-lane lists; seg-0 wave merges 3 partner lists
    // (own lane+16, partner wave lane, partner wave lane+16).
#pragma unroll
    for (int j = 0; j < KNN; ++j) {
        lds_s[wv][j][lane] = ks[j];
        lds_i[wv][j][lane] = ki[j];
    }
    __syncthreads();

    if (active && seg == 0 && lane < 16) {
#pragma unroll
        for (int pidx = 0; pidx < 2 * SPLIT - 1; ++pidx) {
            // partner slots: (wv, lane+16), (wv+1, lane), (wv+1, lane+16)
            const int pwv   = (pidx == 0) ? wv : (wv + 1);
            const int plane = (pidx == 1) ? lane : (lane + 16);
#pragma unroll
            for (int j = 0; j < KNN; ++j) {
                float s = lds_s[pwv][j][plane];
                int   m = lds_i[pwv][j][plane];
                if (s < ks[KNN - 1]) {
                    ks[KNN - 1] = s; ki[KNN - 1] = m;
#pragma unroll
                    for (int jj = KNN - 1; jj > 0; --jj) {
                        if (ks[jj] < ks[jj - 1]) {
                            float ts = ks[jj]; ks[jj] = ks[jj - 1]; ks[jj - 1] = ts;
                            int   ti = ki[jj]; ki[jj] = ki[jj - 1]; ki[jj - 1] = ti;
                        }
                    }
                }
            }
        }

        // ---- epilogue: d2 = score + |p|^2 ; inverse-distance conf weights;
        // signed distance along neighbour normal.
        const float pp = px * px + py * py + pz * pz;
        float acc = 0.0f, wsum = 0.0f;
#pragma unroll
        for (int j = 0; j < KNN; ++j) {
            int   id   = ki[j];
            float d2   = ks[j] + pp;
            float dist = sqrtf(fmaxf(d2, EPS));
            float w    = conf[id] / dist;
            float ccx = cloud[id * 3 + 0];
            float ccy = cloud[id * 3 + 1];
            float ccz = cloud[id * 3 + 2];
            float nx  = normals[id * 3 + 0];
            float ny  = normals[id * 3 + 1];
            float nz  = normals[id * 3 + 2];
            float dp  = (ccx - px) * nx + (ccy - py) * ny + (ccz - pz) * nz;
            float sg  = (dp > 0.0f) ? 1.0f : ((dp < 0.0f) ? -1.0f : 0.0f);
            acc  += dist * sg * w;
            wsum += w;
        }
        out[q] = acc / wsum;
    }
}

extern "C" void kernel_launch(void* const* d_in, const int* in_sizes, int n_in,
                              void* d_out, int out_size, void* d_ws, size_t ws_size,
                              hipStream_t stream) {
    const float* pts     = (const float*)d_in[0];  // [B*N*3]
    const float* cloud   = (const float*)d_in[1];  // [M*3]
    const float* normals = (const float*)d_in[2];  // [M*3]
    const float* conf    = (const float*)d_in[3];  // [M]
    float* out = (float*)d_out;

    const int Q = in_sizes[0] / 3;   // 16384 flat queries
    const int M = in_sizes[3];       // 30000 cloud points

    const int Qtiles = (Q + 15) / 16;
    const int waves  = Qtiles * SPLIT;
    const int blocks = (waves + WPB - 1) / WPB;
    sdf_knn_wmma<<<blocks, 32 * WPB, 0, stream>>>(pts, cloud, normals, conf,
                                                  out, Q, M);
}